// UniGNN_50749333569739
// MI455X (gfx1250) — compile-verified
//
#include <hip/hip_runtime.h>
#include <hip/hip_bf16.h>

// ---------------------------------------------------------------------------
// UniGNN hypergraph-attention layer for MI455X (gfx1250).
// fp32 throughout (matches reference); GEMM via V_WMMA_F32_16X16X4_F32.
// Scatter stages use fp32 global atomics (L2-resident targets: 192MB L2
// holds Xe/denom/amax/out footprints entirely).
// ---------------------------------------------------------------------------

typedef __attribute__((ext_vector_type(2))) float v2f;
typedef __attribute__((ext_vector_type(8))) float v8f;

#define IN_CH 128
#define HC    128   // H*C
#define HH    8
#define CC    16
#define M_EDGES 100000
#define NEG_SLOPE 0.2f

// ---------------------------------------------------------------------------
// Fill buffer with -inf (for segment-max init)
__global__ void fill_neginf(float* __restrict__ p, int n) {
    int g = blockIdx.x * blockDim.x + threadIdx.x;
    if (g < n) p[g] = __int_as_float(0xFF800000u);  // -inf
}

// ---------------------------------------------------------------------------
// X0 = X @ W : full 16-row tiles only, (tiles x 16, 128) @ (128,128).
// One wave computes one 16x16 output tile; 8 waves/block cover all 8
// column tiles of a 16-row strip. EXEC stays all-ones (WMMA requirement),
// no guards anywhere -> clean store path.
//
// VGPR layouts (ISA 7.12.2):
//   A 16x4 f32 : lane l -> M = l%16, Kbase = 2*(l/16); vgpr0=K, vgpr1=K+1
//   B 4x16 f32 : lane l -> N = l%16, Kbase = 2*(l/16); vgpr0=K, vgpr1=K+1
//   C/D 16x16  : lane l -> N = l%16; vgpr r -> M = r + 8*(l/16)
__global__ void gemm_wmma_f32(const float* __restrict__ X,
                              const float* __restrict__ W,
                              float* __restrict__ X0) {
    const int lane = threadIdx.x & 31;
    const int wave = threadIdx.x >> 5;          // 0..7 : column tile
    const int row0 = blockIdx.x * 16;
    const int col0 = wave * 16;
    const int m    = lane & 15;                 // M for A, N for B/D
    const int kb   = (lane >> 4) << 1;          // 0 or 2

    const float* xrow = X + (size_t)(row0 + m) * IN_CH + kb;
    const float* wcol = W + (size_t)kb * HC + (col0 + m);

    v8f acc = {};
    #pragma unroll 8
    for (int k = 0; k < IN_CH; k += 4) {
        v2f a, b;
        a.x = xrow[k];
        a.y = xrow[k + 1];
        b.x = wcol[(size_t)k * HC];
        b.y = wcol[(size_t)k * HC + HC];
        acc = __builtin_amdgcn_wmma_f32_16x16x4_f32(
                  false, a, false, b, (short)0, acc, false, false);
    }

    float* out = X0 + (size_t)(row0 + ((lane >> 4) << 3)) * HC + col0 + m;
    #pragma unroll
    for (int r = 0; r < 8; ++r) {
        out[(size_t)r * HC] = acc[r];
    }
}

// Scalar tail for N % 16 leftover rows (not launched when N % 16 == 0).
__global__ void gemm_tail(const float* __restrict__ X,
                          const float* __restrict__ W,
                          float* __restrict__ X0, int rowStart, int Nrows) {
    int g = blockIdx.x * blockDim.x + threadIdx.x;
    int row = rowStart + (g >> 7);
    int col = g & 127;
    if (row >= Nrows) return;
    const float* xr = X + (size_t)row * IN_CH;
    float acc = 0.0f;
    #pragma unroll 8
    for (int k = 0; k < IN_CH; ++k) acc += xr[k] * W[(size_t)k * HC + col];
    X0[(size_t)row * HC + col] = acc;
}

// ---------------------------------------------------------------------------
// Stage 1: Xe_sum[edge] += X0[vertex] (128 ch), cnt[edge] += 1.
// 32 lanes per incidence, 4 channels (float4) per lane.
__global__ void scatter_to_edges(const float* __restrict__ X0,
                                 const int* __restrict__ vertex,
                                 const int* __restrict__ edges,
                                 float* __restrict__ Xe,
                                 float* __restrict__ cnt, int nnz) {
    size_t g = (size_t)blockIdx.x * blockDim.x + threadIdx.x;
    size_t i = g >> 5;
    int    t = (int)(g & 31);
    if (i >= (size_t)nnz) return;
    int v = vertex[i], e = edges[i];
    const float4 x = *(const float4*)(X0 + (size_t)v * HC + 4 * t);
    float* dst = Xe + (size_t)e * HC + 4 * t;
    atomicAdd(dst + 0, x.x);
    atomicAdd(dst + 1, x.y);
    atomicAdd(dst + 2, x.z);
    atomicAdd(dst + 3, x.w);
    if (t == 0) atomicAdd(cnt + e, 1.0f);
}

// ---------------------------------------------------------------------------
// Stage 2: Xe = Xe_sum / max(cnt,1); la[m,h] = leaky_relu(sum_c Xe*att_e)
// one thread per (edge, head)
__global__ void finalize_edges(float* __restrict__ Xe,
                               const float* __restrict__ cnt,
                               const float* __restrict__ att_e,
                               float* __restrict__ la, int Mh) {
    int g = blockIdx.x * blockDim.x + threadIdx.x;
    if (g >= Mh) return;
    int m = g >> 3, h = g & 7;
    float inv = 1.0f / fmaxf(cnt[m], 1.0f);
    float* xr = Xe + (size_t)m * HC + h * CC;
    const float* ar = att_e + h * CC;
    float dot = 0.0f;
    #pragma unroll
    for (int c = 0; c < CC; ++c) {
        float xv = xr[c] * inv;
        xr[c] = xv;                 // store the mean back in place
        dot += xv * ar[c];
    }
    la[g] = (dot >= 0.0f) ? dot : NEG_SLOPE * dot;
}

// ---------------------------------------------------------------------------
// Float atomic max via signed/unsigned bit trick (valid with -inf init,
// all updates through this function).
__device__ __forceinline__ void atomicMaxFloat(float* addr, float val) {
    if (val >= 0.0f) atomicMax((int*)addr, __float_as_int(val));
    else             atomicMin((unsigned int*)addr, __float_as_uint(val));
}

// Stage 3: amax[vertex,h] = segment_max(la[edges,h])
__global__ void segment_max_k(const float* __restrict__ la,
                              const int* __restrict__ vertex,
                              const int* __restrict__ edges,
                              float* __restrict__ amax, int nnz8) {
    size_t g = (size_t)blockIdx.x * blockDim.x + threadIdx.x;
    if (g >= (size_t)nnz8) return;
    size_t i = g >> 3;
    int    h = (int)(g & 7);
    float val = la[(size_t)edges[i] * HH + h];
    atomicMaxFloat(&amax[(size_t)vertex[i] * HH + h], val);
}

// Stage 4: denom[vertex,h] += exp(la[edges,h] - clamp(amax))
__global__ void segment_denom_k(const float* __restrict__ la,
                                const int* __restrict__ vertex,
                                const int* __restrict__ edges,
                                const float* __restrict__ amax,
                                float* __restrict__ denom, int nnz8) {
    size_t g = (size_t)blockIdx.x * blockDim.x + threadIdx.x;
    if (g >= (size_t)nnz8) return;
    size_t i = g >> 3;
    int    h = (int)(g & 7);
    int v = vertex[i];
    float am = amax[(size_t)v * HH + h];
    if (!(am >= -3.0e38f)) am = 0.0f;           // -inf/NaN -> 0 (isfinite clamp)
    float ex = __expf(la[(size_t)edges[i] * HH + h] - am);
    atomicAdd(&denom[(size_t)v * HH + h], ex);
}

// ---------------------------------------------------------------------------
// Stage 5: out[vertex] += Xe[edge] * alpha;  alpha = ex/(denom+1e-16)
// 32 lanes per incidence, 4 channels per lane (all 4 in one head: t/4).
__global__ void scatter_to_vertices(const float* __restrict__ Xe,
                                    const float* __restrict__ la,
                                    const float* __restrict__ amax,
                                    const float* __restrict__ denom,
                                    const int* __restrict__ vertex,
                                    const int* __restrict__ edges,
                                    float* __restrict__ out, int nnz) {
    size_t g = (size_t)blockIdx.x * blockDim.x + threadIdx.x;
    size_t i = g >> 5;
    int    t = (int)(g & 31);
    if (i >= (size_t)nnz) return;
    int v = vertex[i], e = edges[i], h = t >> 2;
    float am = amax[(size_t)v * HH + h];
    if (!(am >= -3.0e38f)) am = 0.0f;
    float ex    = __expf(la[(size_t)e * HH + h] - am);
    float alpha = ex / (denom[(size_t)v * HH + h] + 1e-16f);
    const float4 x = *(const float4*)(Xe + (size_t)e * HC + 4 * t);
    float* dst = out + (size_t)v * HC + 4 * t;
    atomicAdd(dst + 0, x.x * alpha);
    atomicAdd(dst + 1, x.y * alpha);
    atomicAdd(dst + 2, x.z * alpha);
    atomicAdd(dst + 3, x.w * alpha);
}

// ---------------------------------------------------------------------------
extern "C" void kernel_launch(void* const* d_in, const int* in_sizes, int n_in,
                              void* d_out, int out_size, void* d_ws, size_t ws_size,
                              hipStream_t stream) {
    const float* X     = (const float*)d_in[0];   // (N,128)
    const float* W     = (const float*)d_in[1];   // (128,128)
    const float* att_e = (const float*)d_in[2];   // (1,8,16) = 128
    const int*   vertex = (const int*)d_in[3];    // (nnz,)
    const int*   edges  = (const int*)d_in[4];    // (nnz,)

    const int N   = in_sizes[0] / IN_CH;          // 100000
    const int NNZ = in_sizes[3];                  // 800000
    const int M   = M_EDGES;                      // 100000 (reference constant)

    // Workspace layout (floats); total ~112.4 MB
    float* ws = (float*)d_ws;
    size_t oX0   = 0;
    size_t oXe   = oX0  + (size_t)N * HC;         // edge features (sum -> mean)
    size_t oCnt  = oXe  + (size_t)M * HC;         // edge counts (adjacent to Xe)
    size_t oLa   = oCnt + (size_t)M;              // leaky-relu'd edge logits (M,8)
    size_t oAmax = oLa  + (size_t)M * HH;         // per-vertex max (N,8)
    size_t oDen  = oAmax + (size_t)N * HH;        // per-vertex softmax denom (N,8)
    float* X0    = ws + oX0;
    float* Xe    = ws + oXe;
    float* cnt   = ws + oCnt;
    float* la    = ws + oLa;
    float* amax  = ws + oAmax;
    float* denom = ws + oDen;

    // --- init (graph-capture-safe async memsets + fill kernel) ---
    hipMemsetAsync(Xe, 0, ((size_t)M * HC + (size_t)M) * sizeof(float), stream); // Xe+cnt
    hipMemsetAsync(denom, 0, (size_t)N * HH * sizeof(float), stream);
    hipMemsetAsync(d_out, 0, (size_t)out_size * sizeof(float), stream);
    {
        int n = N * HH;
        fill_neginf<<<(n + 255) / 256, 256, 0, stream>>>(amax, n);
    }

    // --- GEMM: X0 = X @ W (fp32 WMMA on full tiles; scalar tail if needed) ---
    {
        int fullTiles = N / 16;
        if (fullTiles > 0)
            gemm_wmma_f32<<<fullTiles, 256, 0, stream>>>(X, W, X0);
        int tailRows = N - fullTiles * 16;
        if (tailRows > 0) {
            int threads = tailRows * HC;
            gemm_tail<<<(threads + 255) / 256, 256, 0, stream>>>(
                X, W, X0, fullTiles * 16, N);
        }
    }

    // --- mean-aggregate to hyperedges ---
    {
        size_t total = (size_t)NNZ * 32;
        scatter_to_edges<<<(unsigned)((total + 255) / 256), 256, 0, stream>>>(
            X0, vertex, edges, Xe, cnt, NNZ);
    }

    // --- Xe mean + edge attention logits (leaky relu applied per edge) ---
    {
        int n = M * HH;
        finalize_edges<<<(n + 255) / 256, 256, 0, stream>>>(Xe, cnt, att_e, la, n);
    }

    // --- segment softmax over incidences grouped by destination vertex ---
    {
        size_t total = (size_t)NNZ * HH;
        segment_max_k<<<(unsigned)((total + 255) / 256), 256, 0, stream>>>(
            la, vertex, edges, amax, (int)total);
        segment_denom_k<<<(unsigned)((total + 255) / 256), 256, 0, stream>>>(
            la, vertex, edges, amax, denom, (int)total);
    }

    // --- attention-weighted scatter-sum back to vertices ---
    {
        size_t total = (size_t)NNZ * 32;
        scatter_to_vertices<<<(unsigned)((total + 255) / 256), 256, 0, stream>>>(
            Xe, la, amax, denom, vertex, edges, (float*)d_out, NNZ);
    }
}